// HashTopKRouter_45380624449559
// MI455X (gfx1250) — compile-verified
//
#include <hip/hip_runtime.h>
#include <hip/hip_bf16.h>

typedef float v2f __attribute__((ext_vector_type(2)));
typedef float v8f __attribute__((ext_vector_type(8)));

#define N_TOK   16384
#define DIM     2048
#define NE      64
#define KSEL    2
#define NH      2
#define NB      64
#define NS      8
#define CLAMPV  10000.0f

// 16 rows per wave, 4 waves per block => 64 rows per block, 256 blocks.
// Per wave: 4 expert WMMA tiles (16x16 each, covering E=64) + 1 hash tile
// whose N=0,1 columns hold the two hash projections.
__global__ __launch_bounds__(128)
void hash_topk_router_kernel(const float* __restrict__ x,
                             const float* __restrict__ Wr,
                             const float* __restrict__ bias,
                             const float* __restrict__ Wh,
                             const int*  __restrict__ ht,
                             float* __restrict__ out_raw,
                             float* __restrict__ out_logits,
                             float* __restrict__ out_topp,
                             float* __restrict__ out_topi)
{
    __shared__ float s_logits[4][16][NE];   // clamped logits per wave tile
    __shared__ float s_hash[4][16][NH];     // hash logits per wave tile

    const int tid  = threadIdx.x;
    const int w    = tid >> 5;        // wave in block
    const int lane = tid & 31;
    const int n    = lane & 15;       // N (and A-row M) index within tile
    const int kh   = lane >> 4;       // K-half select per ISA 32-bit A/B layout

    const int row_base = blockIdx.x * 64 + w * 16;

    // A: 16x4 f32 tile. lane<16 -> {K0,K1}, lane>=16 -> {K2,K3}, M = lane&15.
    const float* xp  = x  + (size_t)(row_base + n) * DIM + 2 * kh;
    // B: 4x16 f32 tiles of W_router^T: B[k][e] = Wr[e][k].
    const float* wp0 = Wr + (size_t)(0 * 16 + n) * DIM + 2 * kh;
    const float* wp1 = Wr + (size_t)(1 * 16 + n) * DIM + 2 * kh;
    const float* wp2 = Wr + (size_t)(2 * 16 + n) * DIM + 2 * kh;
    const float* wp3 = Wr + (size_t)(3 * 16 + n) * DIM + 2 * kh;
    // Hash B tile: columns 0,1 = W_hash rows; other columns read row 0 (unused garbage).
    const float* hp  = Wh + (size_t)((n < NH) ? n : 0) * DIM + 2 * kh;

    v8f acc0 = {}, acc1 = {}, acc2 = {}, acc3 = {}, acch = {};

    #pragma unroll 4
    for (int k = 0; k < DIM; k += 4) {
        v2f a  = *(const v2f*)(xp  + k);
        v2f b0 = *(const v2f*)(wp0 + k);
        v2f b1 = *(const v2f*)(wp1 + k);
        v2f b2 = *(const v2f*)(wp2 + k);
        v2f b3 = *(const v2f*)(wp3 + k);
        v2f bh = *(const v2f*)(hp  + k);
        acc0 = __builtin_amdgcn_wmma_f32_16x16x4_f32(false, a, false, b0, (short)0, acc0, false, false);
        acc1 = __builtin_amdgcn_wmma_f32_16x16x4_f32(false, a, false, b1, (short)0, acc1, false, false);
        acc2 = __builtin_amdgcn_wmma_f32_16x16x4_f32(false, a, false, b2, (short)0, acc2, false, false);
        acc3 = __builtin_amdgcn_wmma_f32_16x16x4_f32(false, a, false, b3, (short)0, acc3, false, false);
        acch = __builtin_amdgcn_wmma_f32_16x16x4_f32(false, a, false, bh, (short)0, acch, false, false);
    }

    // Bias per expert column handled by this lane in each tile.
    const float bv0 = bias[0 * 16 + n];
    const float bv1 = bias[1 * 16 + n];
    const float bv2 = bias[2 * 16 + n];
    const float bv3 = bias[3 * 16 + n];

    // C/D layout: VGPR r, lanes 0-15 -> M=r ; lanes 16-31 -> M=r+8.
    #pragma unroll
    for (int r = 0; r < 8; ++r) {
        const int    rl   = r + 8 * kh;
        const size_t grow = (size_t)(row_base + rl) * NE;

        float v0 = acc0[r] + bv0;
        float v1 = acc1[r] + bv1;
        float v2 = acc2[r] + bv2;
        float v3 = acc3[r] + bv3;

        out_raw[grow +  0 + n] = v0;
        out_raw[grow + 16 + n] = v1;
        out_raw[grow + 32 + n] = v2;
        out_raw[grow + 48 + n] = v3;

        float c0 = fminf(fmaxf(v0, -CLAMPV), CLAMPV);
        float c1 = fminf(fmaxf(v1, -CLAMPV), CLAMPV);
        float c2 = fminf(fmaxf(v2, -CLAMPV), CLAMPV);
        float c3 = fminf(fmaxf(v3, -CLAMPV), CLAMPV);

        out_logits[grow +  0 + n] = c0;
        out_logits[grow + 16 + n] = c1;
        out_logits[grow + 32 + n] = c2;
        out_logits[grow + 48 + n] = c3;

        s_logits[w][rl][ 0 + n] = c0;
        s_logits[w][rl][16 + n] = c1;
        s_logits[w][rl][32 + n] = c2;
        s_logits[w][rl][48 + n] = c3;

        if (n < NH) s_hash[w][rl][n] = acch[r];
    }

    __syncthreads();

    // Lanes 0-15 each finish one row: bucket -> gather candidates -> top-2 -> softmax.
    if (lane < 16) {
        const int rl   = lane;
        const int grow = row_base + rl;

        int   cand[NH * NS];
        float cs[NH * NS];
        #pragma unroll
        for (int h = 0; h < NH; ++h) {
            float hl = s_hash[w][rl][h];
            float sg = 1.0f / (1.0f + expf(-hl));
            int   b  = (int)floorf(sg * (float)NB);
            b = (b < 0) ? 0 : ((b > NB - 1) ? (NB - 1) : b);
            #pragma unroll
            for (int s = 0; s < NS; ++s) {
                int c = ht[(h * NB + b) * NS + s];
                cand[h * NS + s] = c;
                cs[h * NS + s]   = s_logits[w][rl][c];
            }
        }

        // Stable top-2 (matches lax.top_k tie-breaking: earlier index wins ties).
        float v1 = cs[0]; int p1 = 0;
        float v2 = -__builtin_inff(); int p2 = 0;
        #pragma unroll
        for (int j = 1; j < NH * NS; ++j) {
            float v = cs[j];
            if (v > v1)      { v2 = v1; p2 = p1; v1 = v; p1 = j; }
            else if (v > v2) { v2 = v;  p2 = j; }
        }

        // softmax over (v1, v2), v1 is the max.
        float e2    = expf(v2 - v1);
        float denom = 1.0f + e2 + 1e-12f;
        out_topp[grow * KSEL + 0] = 1.0f / denom;
        out_topp[grow * KSEL + 1] = e2 / denom;
        out_topi[grow * KSEL + 0] = (float)cand[p1];
        out_topi[grow * KSEL + 1] = (float)cand[p2];
    }
}

extern "C" void kernel_launch(void* const* d_in, const int* in_sizes, int n_in,
                              void* d_out, int out_size, void* d_ws, size_t ws_size,
                              hipStream_t stream) {
    (void)in_sizes; (void)n_in; (void)out_size; (void)d_ws; (void)ws_size;

    const float* x    = (const float*)d_in[0];   // (N, D)
    const float* Wr   = (const float*)d_in[1];   // (E, D)
    const float* bias = (const float*)d_in[2];   // (E,)
    const float* Wh   = (const float*)d_in[3];   // (H, D)
    const int*   ht   = (const int*)d_in[4];     // (H, B, S)
    // d_in[5] = return_raw_logits (always produce everything)

    float* out = (float*)d_out;
    float* out_raw    = out;
    float* out_logits = out + (size_t)N_TOK * NE;
    float* out_topp   = out + 2 * (size_t)N_TOK * NE;
    float* out_topi   = out + 2 * (size_t)N_TOK * NE + (size_t)N_TOK * KSEL;

    dim3 grid(N_TOK / 64);   // 256 blocks of 4 waves (64 rows each)
    dim3 block(128);
    hash_topk_router_kernel<<<grid, block, 0, stream>>>(
        x, Wr, bias, Wh, ht, out_raw, out_logits, out_topp, out_topi);
}